// GCN_40037685133533
// MI455X (gfx1250) — compile-verified
//
#include <hip/hip_runtime.h>
#include <hip/hip_bf16.h>

// GCN layer: deg/norm -> h = x@W1 (WMMA f32) -> edge scatter (atomic f32)
// -> out = relu(agg)@W2 + b2 (WMMA f32).
// Sizes are compile-time constants from the reference.

#define N_NODES 100000
#define N_EDGES 1600000
#define F_IN    128
#define HID     128
#define F_OUT   64

typedef __attribute__((ext_vector_type(2))) float v2f;
typedef __attribute__((ext_vector_type(8))) float v8f;

// ---------------- degree / normalization ----------------

__global__ void deg_init_kernel(float* __restrict__ deg) {
  int i = blockIdx.x * blockDim.x + threadIdx.x;
  if (i < N_NODES) deg[i] = 1.0f;  // self-loop contributes weight 1
}

__global__ void deg_accum_kernel(const int* __restrict__ dst,
                                 const float* __restrict__ w,
                                 float* __restrict__ deg) {
  int i = blockIdx.x * blockDim.x + threadIdx.x;
  if (i < N_EDGES)
    __hip_atomic_fetch_add(&deg[dst[i]], w[i],
                           __ATOMIC_RELAXED, __HIP_MEMORY_SCOPE_AGENT);
}

__global__ void dinv_kernel(const float* __restrict__ deg,
                            float* __restrict__ dinv) {
  int i = blockIdx.x * blockDim.x + threadIdx.x;
  if (i < N_NODES) {
    float d = deg[i];
    dinv[i] = (d > 0.0f) ? rsqrtf(d) : 0.0f;
  }
}

// ---------------- GEMM1: h[N,HID] = x[N,F_IN] @ W1[F_IN,HID] ----------------
// One wave per 16x16 output tile. 8 waves/block cover the 8 N-tiles of HID.
// V_WMMA_F32_16X16X4_F32, K-loop of 32 steps.

__global__ void gemm1_kernel(const float* __restrict__ x,
                             const float* __restrict__ W1,
                             float* __restrict__ h) {
  const int lane = threadIdx.x & 31;
  const int wave = threadIdx.x >> 5;
  const int m0   = blockIdx.x * 16;     // N_NODES = 16 * 6250 exactly
  const int n0   = wave * 16;           // HID = 16 * 8
  const int half = lane >> 4;           // K sub-offset selector
  const int l    = lane & 15;

  v8f acc = {};
  // A fragment: lane holds row m0+l, K pair {k+2*half, k+2*half+1} (contiguous)
  const float* arow = x + (size_t)(m0 + l) * F_IN + 2 * half;

  #pragma unroll 8
  for (int k = 0; k < F_IN; k += 4) {
    v2f a = *(const v2f*)(arow + k);                       // 8B load
    v2f b;
    b.x = W1[(size_t)(k + 2 * half + 0) * HID + n0 + l];   // B[k][n]
    b.y = W1[(size_t)(k + 2 * half + 1) * HID + n0 + l];
    acc = __builtin_amdgcn_wmma_f32_16x16x4_f32(
        /*neg_a=*/false, a, /*neg_b=*/false, b,
        /*c_mod=*/(short)0, acc, /*reuse_a=*/false, /*reuse_b=*/false);
  }

  // D layout: acc[r] = D[m0 + 8*half + r][n0 + l]
  float* hp = h + (size_t)(m0 + 8 * half) * HID + n0 + l;
  #pragma unroll
  for (int r = 0; r < 8; ++r) hp[(size_t)r * HID] = acc[r];
}

// ---------------- agg init: bias + self-loop term ----------------
// agg[i][f] = b1[f] + dinv[i]^2 * h[i][f]

__global__ void agg_init_kernel(const float* __restrict__ h,
                                const float* __restrict__ dinv,
                                const float* __restrict__ b1,
                                float* __restrict__ agg) {
  int i = blockIdx.x * blockDim.x + threadIdx.x;  // N_NODES*HID = 12.8M < 2^31
  if (i < N_NODES * HID) {
    int node = i >> 7;      // /HID
    int f    = i & 127;     // %HID
    float di = dinv[node];
    agg[i] = b1[f] + di * di * h[i];
  }
}

// ---------------- edge scatter: agg[dst] += norm * h[src] ----------------
// One wave (32 lanes) per edge; each lane owns 4 consecutive features.
// h and agg both fit in the 192MB L2, so GLOBAL_ATOMIC_ADD_F32 (no-return,
// agent scope) runs at L2 atomic speed.

__global__ void scatter_kernel(const int* __restrict__ src,
                               const int* __restrict__ dst,
                               const float* __restrict__ w,
                               const float* __restrict__ dinv,
                               const float* __restrict__ h,
                               float* __restrict__ agg) {
  int e = blockIdx.x * 8 + (threadIdx.x >> 5);   // 8 waves per 256-thread block
  if (e >= N_EDGES) return;
  int lane = threadIdx.x & 31;

  int s = src[e];
  int d = dst[e];
  float nrm = dinv[s] * w[e] * dinv[d];

  float4 hv = *((const float4*)(h + (size_t)s * HID) + lane);
  float* ap = agg + (size_t)d * HID + lane * 4;
  __hip_atomic_fetch_add(ap + 0, nrm * hv.x, __ATOMIC_RELAXED, __HIP_MEMORY_SCOPE_AGENT);
  __hip_atomic_fetch_add(ap + 1, nrm * hv.y, __ATOMIC_RELAXED, __HIP_MEMORY_SCOPE_AGENT);
  __hip_atomic_fetch_add(ap + 2, nrm * hv.z, __ATOMIC_RELAXED, __HIP_MEMORY_SCOPE_AGENT);
  __hip_atomic_fetch_add(ap + 3, nrm * hv.w, __ATOMIC_RELAXED, __HIP_MEMORY_SCOPE_AGENT);
}

// ---------------- GEMM2: out[N,OUT] = relu(agg) @ W2 + b2 ----------------
// ReLU fused into the A-fragment load; bias fused into the store.
// 8 waves/block: 2 M-tiles x 4 N-tiles (OUT = 64 = 16*4).

__global__ void gemm2_kernel(const float* __restrict__ agg,
                             const float* __restrict__ W2,
                             const float* __restrict__ b2,
                             float* __restrict__ out) {
  const int lane = threadIdx.x & 31;
  const int wave = threadIdx.x >> 5;
  const int m0   = (blockIdx.x * 2 + (wave >> 2)) * 16;
  const int n0   = (wave & 3) * 16;
  const int half = lane >> 4;
  const int l    = lane & 15;

  v8f acc = {};
  const float* arow = agg + (size_t)(m0 + l) * HID + 2 * half;

  #pragma unroll 8
  for (int k = 0; k < HID; k += 4) {
    v2f a;
    a.x = fmaxf(arow[k + 0], 0.0f);   // fused ReLU
    a.y = fmaxf(arow[k + 1], 0.0f);
    v2f b;
    b.x = W2[(size_t)(k + 2 * half + 0) * F_OUT + n0 + l];
    b.y = W2[(size_t)(k + 2 * half + 1) * F_OUT + n0 + l];
    acc = __builtin_amdgcn_wmma_f32_16x16x4_f32(
        false, a, false, b, (short)0, acc, false, false);
  }

  float bias = b2[n0 + l];
  float* op = out + (size_t)(m0 + 8 * half) * F_OUT + n0 + l;
  #pragma unroll
  for (int r = 0; r < 8; ++r) op[(size_t)r * F_OUT] = acc[r] + bias;
}

// ---------------- launcher ----------------

extern "C" void kernel_launch(void* const* d_in, const int* in_sizes, int n_in,
                              void* d_out, int out_size, void* d_ws, size_t ws_size,
                              hipStream_t stream) {
  const float* x  = (const float*)d_in[0];
  const int*   ei = (const int*)d_in[1];   // [2, E]: row0 = src, row1 = dst
  const float* ew = (const float*)d_in[2];
  const float* W1 = (const float*)d_in[3];
  const float* b1 = (const float*)d_in[4];
  const float* W2 = (const float*)d_in[5];
  const float* b2 = (const float*)d_in[6];
  float* out = (float*)d_out;

  const int* src = ei;
  const int* dst = ei + N_EDGES;

  // Workspace layout (floats): deg[Np] | dinv[Np] | h[N*HID] | agg[N*HID]
  const size_t Np = 100096;  // N_NODES rounded up to multiple of 128
  float* deg  = (float*)d_ws;
  float* dinv = deg + Np;
  float* h    = dinv + Np;
  float* agg  = h + (size_t)N_NODES * HID;

  deg_init_kernel <<<(N_NODES + 255) / 256, 256, 0, stream>>>(deg);
  deg_accum_kernel<<<(N_EDGES + 255) / 256, 256, 0, stream>>>(dst, ew, deg);
  dinv_kernel     <<<(N_NODES + 255) / 256, 256, 0, stream>>>(deg, dinv);

  gemm1_kernel    <<<N_NODES / 16, 256, 0, stream>>>(x, W1, h);

  agg_init_kernel <<<(N_NODES * HID + 255) / 256, 256, 0, stream>>>(h, dinv, b1, agg);
  scatter_kernel  <<<(N_EDGES + 7) / 8, 256, 0, stream>>>(src, dst, ew, dinv, h, agg);

  gemm2_kernel    <<<N_NODES / 32, 256, 0, stream>>>(agg, W2, b2, out);
}